// DTILoss_4131758539297
// MI455X (gfx1250) — compile-verified
//
#include <hip/hip_runtime.h>

typedef __attribute__((ext_vector_type(2))) float v2f;
typedef __attribute__((ext_vector_type(8))) float v8f;

#define HW_   65536   // 256*256, power of two
#define HWSH_ 16
#define NB_   64
#define NC_   7
#define NBLOCKS_ 512

// D = R^T R  (R upper-triangular from pred ch1..6); a_i = g_i^T D g_i
// acc = -bval_i * a_i computed as WMMA over K=6 (padded to 8):
//   K order: [D00, D11, D22, D01, D02, D12, 0, 0]
// M = direction (4 chunks of 16), N = pixel (16 consecutive w), per wave32.
__global__ __launch_bounds__(256) void dti_loss_main(
    const float* __restrict__ pred,
    const float* __restrict__ images,
    const float* __restrict__ bvals,
    const float* __restrict__ bvecs,
    float* __restrict__ partial,
    int numTiles)
{
  const int tid   = threadIdx.x;
  const int lane  = tid & 31;
  const int laneM = lane & 15;
  const bool hi   = lane >= 16;

  // ---- per-lane direction coefficients (A-matrix registers), loop-invariant ----
  // A layout (32-bit 16x4): lanes0-15 -> {K0,K1}, lanes16-31 -> {K2,K3}; 2nd WMMA {K4,K5,0,0}
  float a0x[4], a0y[4], a1x[4], a1y[4];
#pragma unroll
  for (int c = 0; c < 4; ++c) {
    int dir  = c * 16 + laneM;
    float gx = bvecs[dir * 3 + 0];
    float gy = bvecs[dir * 3 + 1];
    float gz = bvecs[dir * 3 + 2];
    float nb = -bvals[dir];
    float k0 = nb * gx * gx;          // * D00
    float k1 = nb * gy * gy;          // * D11
    float k2 = nb * gz * gz;          // * D22
    float k3 = 2.0f * nb * gx * gy;   // * D01
    float k4 = 2.0f * nb * gx * gz;   // * D02
    float k5 = 2.0f * nb * gy * gz;   // * D12
    a0x[c] = hi ? k2 : k0;
    a0y[c] = hi ? k3 : k1;
    a1x[c] = hi ? 0.0f : k4;
    a1y[c] = hi ? 0.0f : k5;
  }

  const int waveId = (blockIdx.x * blockDim.x + tid) >> 5;
  const int nWaves = (gridDim.x * blockDim.x) >> 5;

  float lsum = 0.0f;

  for (int tile = waveId; tile < numTiles; tile += nWaves) {
    const int p = (tile << 4) + laneM;       // flat pixel index (b,h,w)
    const int b = p >> HWSH_;
    const int s = p & (HW_ - 1);             // spatial index h*W+w

    const float* pp = pred + (size_t)b * NC_ * HW_ + s;
    const float  S0 = pp[0];
    const float  p1 = pp[(size_t)1 * HW_];
    const float  p2 = pp[(size_t)2 * HW_];
    const float  p3 = pp[(size_t)3 * HW_];
    const float  p4 = pp[(size_t)4 * HW_];
    const float  p5 = pp[(size_t)5 * HW_];
    const float  p6 = pp[(size_t)6 * HW_];

    const float D00 = p1 * p1;
    const float D01 = p1 * p2;
    const float D02 = p1 * p3;
    const float D11 = fmaf(p4, p4, p2 * p2);
    const float D12 = fmaf(p4, p5, p2 * p3);
    const float D22 = fmaf(p6, p6, fmaf(p5, p5, p3 * p3));

    // B-matrix (6x16 padded to 8x16): lanes0-15 -> {K0,K1}, lanes16-31 -> {K2,K3}; 2nd {K4,K5,0,0}
    v2f B0, B1;
    B0.x = hi ? D22 : D00;
    B0.y = hi ? D01 : D11;
    B1.x = hi ? 0.0f : D02;
    B1.y = hi ? 0.0f : D12;

    const float* ib = images + (size_t)b * NB_ * HW_ + s;

    // prefetch next tile's image stream (wave-uniform branch)
    int tnext = tile + nWaves;
    if (tnext < numTiles) {
      const int pn = (tnext << 4) + laneM;
      const int bn = pn >> HWSH_;
      const int sn = pn & (HW_ - 1);
      __builtin_prefetch(images + (size_t)bn * NB_ * HW_ + sn, 0, 0);
    }

#pragma unroll
    for (int c = 0; c < 4; ++c) {
      v2f A0; A0.x = a0x[c]; A0.y = a0y[c];
      v2f A1; A1.x = a1x[c]; A1.y = a1y[c];
      v8f acc = {};
      acc = __builtin_amdgcn_wmma_f32_16x16x4_f32(false, A0, false, B0,
                                                  (short)0, acc, false, false);
      acc = __builtin_amdgcn_wmma_f32_16x16x4_f32(false, A1, false, B1,
                                                  (short)0, acc, false, false);
      // acc[r]: M = c*16 + r (+8 for hi half) = direction, N = laneM = pixel
#pragma unroll
      for (int r = 0; r < 8; ++r) {
        const int dir = c * 16 + r + (hi ? 8 : 0);
        const float img = ib[(size_t)dir * HW_];
        const float pm  = S0 * __expf(acc[r]);
        const float df  = pm - img;
        lsum = fmaf(df, df, lsum);
      }
    }
  }

  // ---- deterministic block reduction ----
  __shared__ float red[256];
  red[tid] = lsum;
  __syncthreads();
#pragma unroll
  for (int off = 128; off > 0; off >>= 1) {
    if (tid < off) red[tid] += red[tid + off];
    __syncthreads();
  }
  if (tid == 0) partial[blockIdx.x] = red[0];
}

__global__ __launch_bounds__(256) void dti_loss_reduce(
    const float* __restrict__ partial, int n, float invN, float* __restrict__ out)
{
  __shared__ float red[256];
  float s = 0.0f;
  for (int i = threadIdx.x; i < n; i += 256) s += partial[i];
  red[threadIdx.x] = s;
  __syncthreads();
#pragma unroll
  for (int off = 128; off > 0; off >>= 1) {
    if (threadIdx.x < off) red[threadIdx.x] += red[threadIdx.x + off];
    __syncthreads();
  }
  if (threadIdx.x == 0) out[0] = red[0] * invN;
}

extern "C" void kernel_launch(void* const* d_in, const int* in_sizes, int n_in,
                              void* d_out, int out_size, void* d_ws, size_t ws_size,
                              hipStream_t stream) {
  const float* pred   = (const float*)d_in[0];
  const float* images = (const float*)d_in[1];
  const float* bvals  = (const float*)d_in[2];
  const float* bvecs  = (const float*)d_in[3];
  float* out = (float*)d_out;
  float* ws  = (float*)d_ws;

  const int numPixels = in_sizes[0] / NC_;     // B*H*W
  const int numTiles  = numPixels >> 4;        // 16 pixels per wave-tile
  const float invN    = 1.0f / (float)in_sizes[1];  // mean over B*Nb*H*W

  dti_loss_main<<<NBLOCKS_, 256, 0, stream>>>(pred, images, bvals, bvecs, ws, numTiles);
  dti_loss_reduce<<<1, 256, 0, stream>>>(ws, NBLOCKS_, invN, out);
}